// add_noise_43868795961991
// MI455X (gfx1250) — compile-verified
//
#include <hip/hip_runtime.h>

// out[b,0,h,w] = clip(A1[b,w] + (A2[b,w] + A3[b,w]^2 + 1) * img[b,0,h,w], 0, 1)
// B=64, C=1, H=1024, W=1024, all float32.
//
// Pure streaming kernel: 256MB in + 256MB out @ 23.3 TB/s => ~23us floor.
// Strategy: float4 (b128) non-temporal loads/stores, per-thread coefficient
// hoisting across 16 rows, full unroll for MLP, one gfx1250 prefetch per
// block to warm GL2 for the next tile.

typedef float v4f __attribute__((ext_vector_type(4)));

#define B_   64
#define H_   1024
#define W_   1024
#define ROWS 16
#define BLOCKS_PER_IMG (H_ / ROWS)          // 64
#define TOTAL_ELEMS ((size_t)B_ * H_ * W_)  // 64M

__global__ __launch_bounds__(256)
void add_noise_43868795961991_kernel(const float* __restrict__ img,
                                     const float* __restrict__ A1,
                                     const float* __restrict__ A2,
                                     const float* __restrict__ A3,
                                     float* __restrict__ out)
{
    const int w4  = threadIdx.x;                 // float4 column index, 0..255
    const int blk = blockIdx.x;                  // 0..4095
    const int b   = blk / BLOCKS_PER_IMG;        // image
    const int h0  = (blk % BLOCKS_PER_IMG) * ROWS;

    // ---- per-(b,w) coefficients: loaded once, reused for 16 rows ----------
    const size_t aoff = (size_t)b * W_ + (size_t)w4 * 4;
    const v4f a1 = *(const v4f*)(A1 + aoff);     // cached (reused by 64 blocks)
    const v4f a2 = *(const v4f*)(A2 + aoff);
    const v4f a3 = *(const v4f*)(A3 + aoff);

    const v4f k = a2 + a3 * a3 + 1.0f;           // gain, hoisted out of H loop

    // ---- streaming section ------------------------------------------------
    const size_t base    = ((size_t)b * H_ + (size_t)h0) * W_ + (size_t)w4 * 4;
    const v4f*   ip      = (const v4f*)(img + base);
    v4f*         op      = (v4f*)(out + base);
    const int    strideV = W_ / 4;               // 256 float4 per row

    // Warm GL2 for the next 16-row tile this wavefront of blocks will touch
    // (gfx1250 global_prefetch_b8; speculative, guarded to stay in-buffer).
    if (base + (size_t)ROWS * W_ < TOTAL_ELEMS) {
        __builtin_prefetch((const char*)(img + base + (size_t)ROWS * W_), 0, 1);
    }

    const v4f kZero = {0.0f, 0.0f, 0.0f, 0.0f};
    const v4f kOne  = {1.0f, 1.0f, 1.0f, 1.0f};

#pragma unroll
    for (int r = 0; r < ROWS; ++r) {
        // img/out are touched exactly once each: non-temporal b128 stream,
        // keeps the 192MB L2 from being thrashed by 512MB of dead lines.
        v4f x = __builtin_nontemporal_load(ip + (size_t)r * strideV);
        v4f y = k * x + a1;                              // v_fma (contracted)
        y = __builtin_elementwise_max(y, kZero);
        y = __builtin_elementwise_min(y, kOne);          // clip(., 0, 1)
        __builtin_nontemporal_store(y, op + (size_t)r * strideV);
    }
}

extern "C" void kernel_launch(void* const* d_in, const int* in_sizes, int n_in,
                              void* d_out, int out_size, void* d_ws, size_t ws_size,
                              hipStream_t stream) {
    const float* img = (const float*)d_in[0];
    const float* A1  = (const float*)d_in[1];
    const float* A2  = (const float*)d_in[2];
    const float* A3  = (const float*)d_in[3];
    float*       out = (float*)d_out;

    dim3 grid(B_ * BLOCKS_PER_IMG);   // 4096 blocks
    dim3 block(256);                  // 8 wave32, one full W row as float4
    add_noise_43868795961991_kernel<<<grid, block, 0, stream>>>(img, A1, A2, A3, out);
}